// BoxRegressionLoss_19207093748290
// MI455X (gfx1250) — compile-verified
//
#include <hip/hip_runtime.h>
#include <stdint.h>

typedef __attribute__((ext_vector_type(2))) float v2f;
typedef __attribute__((ext_vector_type(8))) float v8f;

#define M_PRED 8192
#define N_GT   8192
#define THREADS 1024
#define NWAVES (THREADS / 32)
#define ROWS_PER_BLK 16
#define NBLK (M_PRED / ROWS_PER_BLK)          // 512 stripes
#define COLS_PER_WAVE (N_GT / 32)             // 256 columns per wave
#define TILES_PER_WAVE (COLS_PER_WAVE / 16)   // 16 WMMA tiles per wave
#define MATCH_T2 25.0f                        // (MATCH_THRESH=5)^2

// ---------------------------------------------------------------------------
// Kernel 1: single-workgroup greedy matcher.
//  Phase A (per 16-row stripe): 32 waves compute the 16x8192 squared-distance
//  stripe with V_WMMA_F32_16X16X4_F32:  D = A x B + C, where
//      A(16x4) = (-2px, -2py, -2pz, |p|^2)   per pred row
//      B(4x16) = ( gx ,  gy ,  gz ,  1   )   per gt col
//      C       = |g|^2 broadcast per column
//  so D[m][n] = |p_m - g_n|^2 in a single matrix op per tile.
//  Phase B: 16 sequential masked-argmin passes. Each thread owns 8 contiguous
//  columns (2x b128 loads, masked by one byte of the avail bitmask), reduces a
//  (dist2bits<<32 | col) key via wave shuffle-min, then wave0 combines the 32
//  per-wave minima. Only 2 barriers per row.
// ---------------------------------------------------------------------------
__global__ __launch_bounds__(THREADS)
void greedy_match_kernel(const float* __restrict__ pred,
                         const float* __restrict__ gt,
                         float* __restrict__ dist_blk,   // [16][N_GT] scratch
                         int* __restrict__ midx,
                         int* __restrict__ mok)
{
    __shared__ uint32_t avail[N_GT / 32];   // 256 words = 1 KB
    __shared__ uint64_t swmin[NWAVES];      // 32 * 8 B

    const int tid  = threadIdx.x;
    const int lane = tid & 31;
    const int wave = tid >> 5;
    const int half = lane >> 4;             // 0: K=0,1   1: K=2,3 (A/B layout)
    const int l15  = lane & 15;

    for (int w = tid; w < N_GT / 32; w += THREADS) avail[w] = 0xFFFFFFFFu;
    __syncthreads();

    for (int blk = 0; blk < NBLK; ++blk) {
        // ---- Phase A: WMMA squared-distance stripe -------------------------
        {
            const int   m   = blk * ROWS_PER_BLK + l15;
            const float px  = pred[m * 7 + 0];
            const float py  = pred[m * 7 + 1];
            const float pz  = pred[m * 7 + 2];
            const float pn2 = px * px + py * py + pz * pz;
            v2f a;
            if (half == 0) { a.x = -2.0f * px; a.y = -2.0f * py; }
            else           { a.x = -2.0f * pz; a.y = pn2; }

            const int cbase = wave * COLS_PER_WAVE;
            for (int t = 0; t < TILES_PER_WAVE; ++t) {
                const int n = cbase + t * 16 + l15;
                if (t + 1 < TILES_PER_WAVE)
                    __builtin_prefetch(&gt[(n + 16) * 7], 0, 3);  // global_prefetch_b8

                const float gx  = gt[n * 7 + 0];
                const float gy  = gt[n * 7 + 1];
                const float gz  = gt[n * 7 + 2];
                const float gn2 = gx * gx + gy * gy + gz * gz;

                v2f b;
                if (half == 0) { b.x = gx; b.y = gy; }
                else           { b.x = gz; b.y = 1.0f; }

                v8f c;
                #pragma unroll
                for (int k = 0; k < 8; ++k) c[k] = gn2;   // C[m][n] = |g_n|^2

                v8f d = __builtin_amdgcn_wmma_f32_16x16x4_f32(
                            /*neg_a=*/false, a, /*neg_b=*/false, b,
                            /*c_mod=*/(short)0, c,
                            /*reuse_a=*/false, /*reuse_b=*/false);

                const int rbase = half * 8;   // lanes 16-31 hold rows 8..15
                #pragma unroll
                for (int k = 0; k < 8; ++k) {
                    const float v = d[k];
                    dist_blk[(rbase + k) * N_GT + n] = (v > 0.0f) ? v : 0.0f;
                }
            }
        }
        __threadfence_block();
        __syncthreads();

        // ---- Phase B: sequential greedy over the 16 rows -------------------
        for (int r = 0; r < ROWS_PER_BLK; ++r) {
            const float4* drow4 =
                (const float4*)(dist_blk + (size_t)r * N_GT);
            const int j0 = tid * 8;                       // 8 contiguous cols
            const uint32_t am = (avail[j0 >> 5] >> (j0 & 31)) & 0xFFu;

            const float4 va = drow4[tid * 2 + 0];         // global_load_b128
            const float4 vb = drow4[tid * 2 + 1];         // global_load_b128

            uint64_t best = ~0ull;
            const float vals[8] = { va.x, va.y, va.z, va.w,
                                    vb.x, vb.y, vb.z, vb.w };
            #pragma unroll
            for (int o = 0; o < 8; ++o) {
                if (am & (1u << o)) {
                    // dist^2 >= 0 -> float bits order-preserving; low 32 bits
                    // carry the column so min() == first-tie-wins argmin.
                    const uint64_t key =
                        ((uint64_t)__float_as_uint(vals[o]) << 32) |
                        (uint32_t)(j0 + o);
                    if (key < best) best = key;
                }
            }

            // wave-level 64-bit min (lane permutes, no barrier)
            #pragma unroll
            for (int msk = 16; msk >= 1; msk >>= 1) {
                const uint64_t o =
                    (uint64_t)__shfl_xor((unsigned long long)best, msk, 32);
                if (o < best) best = o;
            }
            if (lane == 0) swmin[wave] = best;
            __syncthreads();

            if (wave == 0) {
                uint64_t b2 = swmin[lane];
                #pragma unroll
                for (int msk = 16; msk >= 1; msk >>= 1) {
                    const uint64_t o =
                        (uint64_t)__shfl_xor((unsigned long long)b2, msk, 32);
                    if (o < b2) b2 = o;
                }
                if (lane == 0) {
                    const int   j  = (int)(uint32_t)b2;
                    const float d2 = __uint_as_float((uint32_t)(b2 >> 32));
                    const int   ok = (d2 < MATCH_T2) ? 1 : 0;
                    if (ok) avail[j >> 5] &= ~(1u << (j & 31));
                    midx[blk * ROWS_PER_BLK + r] = ok ? j : 0;
                    mok [blk * ROWS_PER_BLK + r] = ok;
                }
            }
            __syncthreads();   // avail update visible before next row
        }
    }
}

// ---------------------------------------------------------------------------
// Kernel 2: per-pred loss terms + deterministic block reduction -> partials
// ---------------------------------------------------------------------------
__device__ __forceinline__ float smooth_l1(float x) {
    const float a = fabsf(x);
    return (a < 1.0f) ? 0.5f * a * a : a - 0.5f;
}

__global__ __launch_bounds__(256)
void loss_partial_kernel(const float* __restrict__ pred,
                         const float* __restrict__ gt,
                         const int* __restrict__ midx,
                         const int* __restrict__ mok,
                         float* __restrict__ partials)   // [gridDim.x][5]
{
    __shared__ float red[5][256];
    const int tid = threadIdx.x;
    const int i   = blockIdx.x * 256 + tid;

    float sc = 0.0f, ss = 0.0f, so = 0.0f, si = 0.0f, cnt = 0.0f;
    if (mok[i]) {
        const float* p = pred + (size_t)i * 7;
        const float* g = gt   + (size_t)midx[i] * 7;
        cnt = 1.0f;
        sc = smooth_l1(p[0] - g[0]) + smooth_l1(p[1] - g[1]) + smooth_l1(p[2] - g[2]);
        ss = smooth_l1(p[3] - g[3]) + smooth_l1(p[4] - g[4]) + smooth_l1(p[5] - g[5]);
        float dth = p[6] - g[6];
        dth = atan2f(sinf(dth), cosf(dth));
        so = smooth_l1(dth);
        // 2D IoU on (x, y, l, w)
        const float x1 = p[0], y1 = p[1], l1 = p[3], w1 = p[4];
        const float x2 = g[0], y2 = g[1], l2 = g[3], w2 = g[4];
        float iw = fminf(x1 + 0.5f * l1, x2 + 0.5f * l2) -
                   fmaxf(x1 - 0.5f * l1, x2 - 0.5f * l2);
        float ih = fminf(y1 + 0.5f * w1, y2 + 0.5f * w2) -
                   fmaxf(y1 - 0.5f * w1, y2 - 0.5f * w2);
        iw = fmaxf(iw, 0.0f);
        ih = fmaxf(ih, 0.0f);
        const float inter = iw * ih;
        const float uni   = l1 * w1 + l2 * w2 - inter;
        si = 1.0f - inter / (uni + 1e-6f);
    }
    red[0][tid] = sc; red[1][tid] = ss; red[2][tid] = so;
    red[3][tid] = si; red[4][tid] = cnt;
    __syncthreads();
    for (int s = 128; s > 0; s >>= 1) {
        if (tid < s) {
            #pragma unroll
            for (int q = 0; q < 5; ++q) red[q][tid] += red[q][tid + s];
        }
        __syncthreads();
    }
    if (tid < 5) partials[blockIdx.x * 5 + tid] = red[tid][0];
}

// ---------------------------------------------------------------------------
// Kernel 3: finalize (deterministic serial sum of 32 partials)
// ---------------------------------------------------------------------------
__global__ void finalize_kernel(const float* __restrict__ partials,
                                float* __restrict__ out, int nblocks)
{
    float s0 = 0, s1 = 0, s2 = 0, s3 = 0, s4 = 0;
    for (int b = 0; b < nblocks; ++b) {
        s0 += partials[b * 5 + 0];
        s1 += partials[b * 5 + 1];
        s2 += partials[b * 5 + 2];
        s3 += partials[b * 5 + 3];
        s4 += partials[b * 5 + 4];
    }
    const float k = fmaxf(s4, 1.0f);
    // W_CENTER=1.0, W_SIZE=0.5, W_IOU=2.0
    const float loss = 1.0f * (s0 / (3.0f * k)) +
                       0.5f * ((s1 / (3.0f * k)) + s2 / k) +
                       2.0f * (s3 / k);
    out[0] = loss;
}

// ---------------------------------------------------------------------------
extern "C" void kernel_launch(void* const* d_in, const int* in_sizes, int n_in,
                              void* d_out, int out_size, void* d_ws, size_t ws_size,
                              hipStream_t stream)
{
    (void)in_sizes; (void)n_in; (void)out_size; (void)ws_size;
    const float* pred = (const float*)d_in[0];   // [8192][7]
    const float* gt   = (const float*)d_in[1];   // [8192][7]

    char*  ws       = (char*)d_ws;
    float* dist_blk = (float*)ws;                                  // 512 KB
    int*   midx     = (int*)(ws + (size_t)ROWS_PER_BLK * N_GT * 4);
    int*   mok      = (int*)(ws + (size_t)ROWS_PER_BLK * N_GT * 4 + (size_t)M_PRED * 4);
    float* partials = (float*)(ws + (size_t)ROWS_PER_BLK * N_GT * 4 + (size_t)2 * M_PRED * 4);

    greedy_match_kernel<<<1, THREADS, 0, stream>>>(pred, gt, dist_blk, midx, mok);
    loss_partial_kernel<<<M_PRED / 256, 256, 0, stream>>>(pred, gt, midx, mok, partials);
    finalize_kernel<<<1, 1, 0, stream>>>(partials, (float*)d_out, M_PRED / 256);
}